// BufCache_23141283790870
// MI455X (gfx1250) — compile-verified
//
#include <hip/hip_runtime.h>
#include <hip/hip_bf16.h>

// Sliding-window KV cache update for MI455X (gfx1250).
// Pure streaming copy: widest vector path (b128), non-temporal hints,
// no LDS round trip. ~512 MiB total traffic -> ~22us at 23.3 TB/s.

typedef __attribute__((ext_vector_type(4))) unsigned int v4u;  // 16 bytes = 8 bf16

static constexpr int  MAX_LEN  = 8192;
static constexpr int  NEW_LEN  = 16;
static constexpr int  HEAD     = 128;                 // bf16 per row
static constexpr int  VPR      = HEAD * 2 / 16;       // 16 x 16B vectors per row
static constexpr int  PLANES   = 2 * 32;              // batch * heads
static constexpr long long PLANE_VEC = (long long)MAX_LEN * VPR;   // 131072 vecs/plane
static constexpr long long TOT_VEC   = (long long)PLANES * PLANE_VEC;
static constexpr int  KEEP_ROWS = MAX_LEN - NEW_LEN;               // 8176
static constexpr long long KEEP_VEC  = (long long)KEEP_ROWS * VPR; // 130816 = 511*256
static constexpr long long SHIFT_VEC = (long long)NEW_LEN * VPR;   // 256

// Bulk: out[plane][0 .. 8176*16) = cache[plane][16*16 .. 8192*16)
// grid.x = 511 (KEEP_VEC/256), grid.y = plane (64), grid.z = 0:K / 1:V
__global__ __launch_bounds__(256)
void slide_bulk_kernel(const v4u* __restrict__ cache_k,
                       const v4u* __restrict__ cache_v,
                       v4u* __restrict__ out_k,
                       v4u* __restrict__ out_v)
{
    const v4u* __restrict__ src = blockIdx.z ? cache_v : cache_k;
    v4u*       __restrict__ dst = blockIdx.z ? out_v   : out_k;

    const long long base = (long long)blockIdx.y * PLANE_VEC;
    const long long o    = (long long)blockIdx.x * 256 + threadIdx.x;  // < KEEP_VEC

    v4u val = __builtin_nontemporal_load(src + base + SHIFT_VEC + o);
    __builtin_nontemporal_store(val, dst + base + o);
}

// Tail: out[plane][8176*16 .. 8192*16) = new[plane][0 .. 256)
// grid.x = plane (64), grid.z = 0:K / 1:V, 256 threads = 16 rows x 16 vecs
__global__ __launch_bounds__(256)
void slide_tail_kernel(const v4u* __restrict__ k_new,
                       const v4u* __restrict__ v_new,
                       v4u* __restrict__ out_k,
                       v4u* __restrict__ out_v)
{
    const v4u* __restrict__ src = blockIdx.z ? v_new : k_new;
    v4u*       __restrict__ dst = blockIdx.z ? out_v : out_k;

    const long long plane = blockIdx.x;
    const int t = threadIdx.x;  // 0..255

    v4u val = __builtin_nontemporal_load(src + plane * SHIFT_VEC + t);
    __builtin_nontemporal_store(val, dst + plane * PLANE_VEC + KEEP_VEC + t);
}

extern "C" void kernel_launch(void* const* d_in, const int* in_sizes, int n_in,
                              void* d_out, int out_size, void* d_ws, size_t ws_size,
                              hipStream_t stream) {
    (void)in_sizes; (void)n_in; (void)out_size; (void)d_ws; (void)ws_size;

    const v4u* cache_k = (const v4u*)d_in[0];
    const v4u* cache_v = (const v4u*)d_in[1];
    const v4u* k_new   = (const v4u*)d_in[2];
    const v4u* v_new   = (const v4u*)d_in[3];

    v4u* out_k = (v4u*)d_out;
    v4u* out_v = out_k + TOT_VEC;   // tuple (new_k, new_v) packed flat in order

    // Bulk shifted copy: 511 * 64 * 2 blocks of 256 threads, one b128 vec each.
    dim3 bulk_grid((unsigned)(KEEP_VEC / 256), PLANES, 2);
    slide_bulk_kernel<<<bulk_grid, 256, 0, stream>>>(cache_k, cache_v, out_k, out_v);

    // Tail: append the 16 new rows per plane.
    dim3 tail_grid(PLANES, 1, 2);
    slide_tail_kernel<<<tail_grid, 256, 0, stream>>>(k_new, v_new, out_k, out_v);
}